// Net_44890998178164
// MI455X (gfx1250) — compile-verified
//
#include <hip/hip_runtime.h>

typedef __attribute__((ext_vector_type(2))) float v2f;
typedef __attribute__((ext_vector_type(4))) float v4f;
typedef __attribute__((ext_vector_type(8))) float v8f;

#define VOCAB  128
#define HIDDEN 128

// ---------------------------------------------------------------------------
// Kernel 1: T_t = emb @ W_t^T  for t in {0,1}, using V_WMMA_F32_16X16X4_F32.
//   T[t][v][o] = sum_h emb[v][h] * W[o][t*128 + h]
// One wave (32 threads) computes one 16x16 tile of T_t.
// Grid: (8, 8, 2) tiles; K=128 consumed 4 at a time (32 WMMAs per wave).
//
// Register layouts per CDNA5 ISA 7.12.2:
//  A (16x4 f32, M x K): lane l<16 -> row M=l, v0=K0, v1=K1;
//                       lane l>=16 -> row M=l-16, v0=K2, v1=K3.
//  B (4x16 f32, K x N): lane l<16 -> col N=l, v0=K0, v1=K1;
//                       lane l>=16 -> col N=l-16, v0=K2, v1=K3.
//  C/D (16x16 f32):     VGPR r -> M=r (lanes 0-15, N=lane)
//                               / M=r+8 (lanes 16-31, N=lane-16).
// ---------------------------------------------------------------------------
__global__ void precompute_tables_wmma(const float* __restrict__ emb,   // [128,128]
                                       const float* __restrict__ W,     // [128,256]
                                       float* __restrict__ T)           // [2,128,128]
{
    const int lane = threadIdx.x;          // 0..31, full wave, EXEC all ones
    const int half = lane >> 4;            // 0: K pair {0,1}, 1: K pair {2,3}
    const int l    = lane & 15;

    const int i = blockIdx.x;              // row tile   (vocab)
    const int j = blockIdx.y;              // col tile   (out features)
    const int t = blockIdx.z;              // which half of W

    v8f acc = {};

    const float* arow = emb + (16 * i + l) * HIDDEN + 2 * half;            // emb[v][k..]
    const float* brow = W   + (16 * j + l) * (2 * HIDDEN) + t * HIDDEN + 2 * half; // W[o][t*128+k..]

#pragma unroll 4
    for (int k = 0; k < HIDDEN; k += 4) {
        v2f a = { arow[k],     arow[k + 1] };
        v2f b = { brow[k],     brow[k + 1] };
        // 8 args: (neg_a, A, neg_b, B, c_mod, C, reuse_a, reuse_b)
        acc = __builtin_amdgcn_wmma_f32_16x16x4_f32(
            false, a, false, b, (short)0, acc, false, false);
    }

    // Scatter D back: element (m0+r, l) of this 16x16 tile.
    const int m0 = 8 * half;
    float* dst = T + (size_t)t * VOCAB * HIDDEN + (size_t)(16 * i + m0) * HIDDEN + 16 * j + l;
#pragma unroll
    for (int r = 0; r < 8; ++r)
        dst[(size_t)r * HIDDEN] = acc[r];
}

// ---------------------------------------------------------------------------
// Kernel 2: out[e][:] = T1[z[src_e]][:] + T2[z[dst_e]][:] + b[:]
// One wave per edge; each lane owns one float4 (32 lanes x 4 = 128 floats).
// Pure streaming: ~256 MB of nontemporal stores -> HBM-bound (~11 us @ 23.3 TB/s).
// Tables (128 KB) + z (400 KB) stay L2-resident.
// ---------------------------------------------------------------------------
__global__ void edge_gather_add(const int*   __restrict__ z,     // [N]
                                const int*   __restrict__ ei,    // [2,E] row-major
                                const float* __restrict__ T,     // [2,128,128]
                                const float* __restrict__ bias,  // [128]
                                float*       __restrict__ out,   // [E,128]
                                int E)
{
    const int gtid = blockIdx.x * blockDim.x + threadIdx.x;
    const int e    = gtid >> 5;
    const int lane = threadIdx.x & 31;
    if (e >= E) return;

    const int s = __builtin_nontemporal_load(ei + e);        // src node
    const int d = __builtin_nontemporal_load(ei + E + e);    // dst node
    const int vs = z[s];
    const int vd = z[d];

    const v4f* t1 = (const v4f*)(T + (size_t)vs * HIDDEN);
    const v4f* t2 = (const v4f*)(T + (size_t)VOCAB * HIDDEN + (size_t)vd * HIDDEN);
    const v4f* bb = (const v4f*)bias;

    v4f r = t1[lane] + t2[lane] + bb[lane];

    __builtin_nontemporal_store(r, (v4f*)(out + (size_t)e * HIDDEN) + lane);
}

// ---------------------------------------------------------------------------
// Inputs (setup_inputs order):
//   d_in[0] z          int32  [100000]
//   d_in[1] edge_index int32  [2,500000]
//   d_in[2] emb_table  f32    [128,128]
//   d_in[3] W          f32    [128,256]
//   d_in[4] b          f32    [128]
// Output: f32 [500000,128] (trailing 1,1 dims flat)
// Workspace: 2*128*128*4 = 128 KB for T1/T2.
// ---------------------------------------------------------------------------
extern "C" void kernel_launch(void* const* d_in, const int* in_sizes, int n_in,
                              void* d_out, int out_size, void* d_ws, size_t ws_size,
                              hipStream_t stream) {
    const int*   z    = (const int*)d_in[0];
    const int*   ei   = (const int*)d_in[1];
    const float* emb  = (const float*)d_in[2];
    const float* W    = (const float*)d_in[3];
    const float* bias = (const float*)d_in[4];
    float*       out  = (float*)d_out;
    float*       T    = (float*)d_ws;            // [2][128][128]

    const int E = in_sizes[1] / 2;

    dim3 gridT(VOCAB / 16, HIDDEN / 16, 2);      // (8,8,2) tiles
    precompute_tables_wmma<<<gridT, 32, 0, stream>>>(emb, W, T);

    const int wavesPerBlock = 8;                 // 256 threads = 8 wave32
    const int blocks = (E + wavesPerBlock - 1) / wavesPerBlock;
    edge_gather_add<<<blocks, 256, 0, stream>>>(z, ei, T, bias, out, E);
}